// State_selector_90907277787365
// MI455X (gfx1250) — compile-verified
//
#include <hip/hip_runtime.h>
#include <hip/hip_bf16.h>
#include <math.h>

#define B 64
#define S 64
#define H 2048

typedef __attribute__((ext_vector_type(2))) float v2f;
typedef __attribute__((ext_vector_type(8))) float v8f;

// ---------------------------------------------------------------------------
// Kernel 1: pred = tanh(prev_state @ w1) using V_WMMA_F32_16X16X4_F32.
// Grid: (N/128, M/16) blocks of 256 threads (8 waves). Each wave owns one
// 16x16 tile of the 64x2048 output, stepping K by 4 per WMMA.
//
// 32-bit A 16x4 layout : lanes 0-15 -> M=lane,    VGPR0=K0, VGPR1=K1
//                        lanes16-31 -> M=lane-16, VGPR0=K2, VGPR1=K3
// 32-bit B 4x16 layout : lanes 0-15 -> N=lane,    VGPR0=K0, VGPR1=K1
//                        lanes16-31 -> N=lane-16, VGPR0=K2, VGPR1=K3
// 32-bit C/D 16x16     : VGPR v: lanes 0-15 -> M=v, lanes 16-31 -> M=v+8
// ---------------------------------------------------------------------------
__global__ __launch_bounds__(256) void gemm_tanh_wmma(
    const float* __restrict__ prev,  // [B, H]
    const float* __restrict__ w1,    // [H, H]
    float* __restrict__ pred)        // [B, H] workspace
{
  const int lane = threadIdx.x & 31;
  const int wave = threadIdx.x >> 5;
  const int half = lane >> 4;   // 0: lanes 0-15, 1: lanes 16-31
  const int r    = lane & 15;

  const int m0 = blockIdx.y * 16;               // row tile base (0..48)
  const int n0 = blockIdx.x * 128 + wave * 16;  // col tile base

  const float* __restrict__ arow = prev + (size_t)(m0 + r) * H;

  v8f acc = {};
  for (int k = 0; k < H; k += 4) {
    const int ka = k + 2 * half;
    v2f a, b;
    a.x = arow[ka];
    a.y = arow[ka + 1];
    b.x = w1[(size_t)ka * H + n0 + r];
    b.y = w1[(size_t)(ka + 1) * H + n0 + r];
    acc = __builtin_amdgcn_wmma_f32_16x16x4_f32(
        /*neg_a=*/false, a, /*neg_b=*/false, b,
        /*c_mod=*/(short)0, acc, /*reuse_a=*/false, /*reuse_b=*/false);
  }

  const int col = n0 + r;
#pragma unroll
  for (int v = 0; v < 8; ++v) {
    const int row = m0 + v + 8 * half;
    pred[(size_t)row * H + col] = tanhf(acc[v]);
  }
}

// ---------------------------------------------------------------------------
// Kernel 2: per-row L2 of (pred - new_state), replacement mask, and stable
// argsort rank of the modified norm vector. One block per batch row.
// rank(s) = destination slot of source slot s under jnp.argsort(norms).
// ---------------------------------------------------------------------------
__global__ __launch_bounds__(256) void norm_rank_kernel(
    const float* __restrict__ pred,       // [B, H]
    const float* __restrict__ new_state,  // [B, H]
    const float* __restrict__ old_norms,  // [B, S]
    float* __restrict__ out_l2,           // d_out tail, [B]
    float* __restrict__ l2_ws,            // [B]
    int* __restrict__ perm_ws,            // [B, S]
    int* __restrict__ mask_ws)            // [B]
{
  __shared__ float sdata[256];
  __shared__ float norms[S];
  __shared__ float l2_sh;
  __shared__ int mask_sh;

  const int b   = blockIdx.x;
  const int tid = threadIdx.x;

  // --- sum of squared diffs over H, strided ---
  float acc = 0.0f;
  const float* p = pred + (size_t)b * H;
  const float* q = new_state + (size_t)b * H;
  for (int h = tid; h < H; h += 256) {
    const float d = p[h] - q[h];
    acc += d * d;
  }
  sdata[tid] = acc;
  __syncthreads();
#pragma unroll
  for (int off = 128; off > 0; off >>= 1) {
    if (tid < off) sdata[tid] += sdata[tid + off];
    __syncthreads();
  }

  if (tid == 0) {
    const float l2 = sqrtf(sdata[0]);
    l2_sh = l2;
    l2_ws[b] = l2;
    out_l2[b] = l2;
    mask_sh = (l2 > old_norms[(size_t)b * S]) ? 1 : 0;
    mask_ws[b] = mask_sh;
  }
  __syncthreads();

  // --- build modified norm vector ---
  if (tid < S) {
    float v = old_norms[(size_t)b * S + tid];
    if (tid == 0 && mask_sh) v = l2_sh;
    norms[tid] = v;
  }
  __syncthreads();

  // --- stable rank: destination position of each source slot ---
  if (tid < S) {
    const float ns = norms[tid];
    int rank = 0;
#pragma unroll 8
    for (int t = 0; t < S; ++t) {
      const float nt = norms[t];
      rank += (nt < ns || (nt == ns && t < tid)) ? 1 : 0;
    }
    perm_ws[b * S + tid] = rank;
  }
}

// ---------------------------------------------------------------------------
// Kernel 3: permuted streaming copy of the 33.5 MB state buffer.
// One block per (b, s): 2048 floats = 512 float4 -> 2 float4 per thread.
// Slot 0 conditionally sourced from new_state.
// ---------------------------------------------------------------------------
__global__ __launch_bounds__(256) void scatter_states_kernel(
    const float* __restrict__ old_states,  // [B, S, H]
    const float* __restrict__ new_state,   // [B, H]
    const int* __restrict__ perm_ws,       // [B, S]
    const int* __restrict__ mask_ws,       // [B]
    float* __restrict__ out_states)        // [B, S, H]
{
  const int bs = blockIdx.x;
  const int b  = bs >> 6;   // /S
  const int s  = bs & 63;   // %S

  const float* src;
  if (s == 0) {
    src = mask_ws[b] ? (new_state + (size_t)b * H)
                     : (old_states + (size_t)b * S * H);
  } else {
    src = old_states + ((size_t)b * S + s) * H;
  }
  const int dslot = perm_ws[b * S + s];
  float* dst = out_states + ((size_t)b * S + dslot) * H;

  const float4* __restrict__ s4 = (const float4*)src;
  float4* __restrict__ d4 = (float4*)dst;
  const int tid = threadIdx.x;
#pragma unroll
  for (int i = 0; i < 2; ++i) {
    const int idx = tid + i * 256;  // 512 float4 total
    d4[idx] = s4[idx];
  }
}

extern "C" void kernel_launch(void* const* d_in, const int* in_sizes, int n_in,
                              void* d_out, int out_size, void* d_ws, size_t ws_size,
                              hipStream_t stream) {
  const float* old_states = (const float*)d_in[0];  // [B,S,H]
  const float* new_state  = (const float*)d_in[1];  // [B,H]
  const float* prev_state = (const float*)d_in[2];  // [B,H]
  const float* old_norms  = (const float*)d_in[3];  // [B,S]
  const float* w1         = (const float*)d_in[4];  // [H,H]

  float* out_states = (float*)d_out;                 // [B,S,H]
  float* out_l2     = (float*)d_out + (size_t)B * S * H;  // [B]

  // workspace layout
  float* pred  = (float*)d_ws;                       // B*H floats
  float* l2_ws = pred + (size_t)B * H;               // B floats
  int*   perm  = (int*)(l2_ws + B);                  // B*S ints
  int*   maskw = perm + B * S;                       // B ints

  // K1: GEMM + tanh. grid covers 2048 cols / 128 per block, 64 rows / 16.
  dim3 g1(H / 128, B / 16);
  gemm_tanh_wmma<<<g1, 256, 0, stream>>>(prev_state, w1, pred);

  // K2: per-row L2 + mask + stable argsort ranks.
  norm_rank_kernel<<<B, 256, 0, stream>>>(pred, new_state, old_norms,
                                          out_l2, l2_ws, perm, maskw);

  // K3: permuted copy of all state slots.
  scatter_states_kernel<<<B * S, 256, 0, stream>>>(old_states, new_state,
                                                   perm, maskw, out_states);
}